// DetectHead_2997887173049
// MI455X (gfx1250) — compile-verified
//
#include <hip/hip_runtime.h>
#include <stdint.h>

#define B_        16
#define C_        80
#define NANCH     17064
#define SORTN     32768            // next pow2 >= NANCH, for bitonic sort
#define CHUNK     2048             // elements sorted/merged per block in LDS
#define NTOP      1000
#define SCORE_THR 0.05f
#define IOU_THR   0.6f

typedef __attribute__((ext_vector_type(4))) unsigned int u32x4;
typedef __attribute__((ext_vector_type(4))) int          i32x4;
typedef __attribute__((ext_vector_type(8))) int          i32x8;

struct Feats {
  const float* cls[5];
  const float* ctr[5];
  const float* reg[5];
};

__device__ __forceinline__ float sigmoidf_(float x) {
  return 1.0f / (1.0f + expf(-x));
}

// monotone float -> uint32 key (order preserving)
__device__ __forceinline__ uint32_t fkey(float f) {
  uint32_t u = __float_as_uint(f);
  return u ^ ((u >> 31) ? 0xFFFFFFFFu : 0x80000000u);
}
__device__ __forceinline__ float unfkey(uint32_t u) {
  uint32_t v = u ^ ((u & 0x80000000u) ? 0x80000000u : 0xFFFFFFFFu);
  return __uint_as_float(v);
}

// gfx1250 async global<->LDS (ASYNCcnt path). Low 32 bits of a flat shared
// pointer are the wave-relative LDS byte address (aperture layout, ISA 10.2).
__device__ __forceinline__ void async_g2l_b128(void* lds, const void* gsrc) {
  uint32_t ldsoff = (uint32_t)(uintptr_t)lds;
  asm volatile("global_load_async_to_lds_b128 %0, %1, off"
               :: "v"(ldsoff), "v"(gsrc) : "memory");
}
__device__ __forceinline__ void async_l2g_b128(void* gdst, const void* lds) {
  uint32_t ldsoff = (uint32_t)(uintptr_t)lds;
  asm volatile("global_store_async_from_lds_b128 %0, %1, off"
               :: "v"(gdst), "v"(ldsoff) : "memory");
}
__device__ __forceinline__ void wait_async0() {
  asm volatile("s_wait_asynccnt 0" ::: "memory");
}

// ---------------------------------------------------------------------------
// Kernel 1: per-anchor score (sigmoid-max over 80 classes * sigmoid(ctr)),
// argmax class, box decode, packed sort key (+ pad keys).
// ---------------------------------------------------------------------------
__global__ void k_score(Feats f, uint64_t* __restrict__ keys,
                        float* __restrict__ boxes_all, int* __restrict__ cls_all) {
  int t = blockIdx.x * blockDim.x + threadIdx.x;
  if (t >= B_ * SORTN) return;
  int b = t / SORTN;
  int n = t - b * SORTN;
  uint64_t* kb = keys + (size_t)b * SORTN;
  if (n >= NANCH) { kb[n] = 0ull; return; }  // pad: sorts to the end (descending)

  int p, w, s, hw, l;
  if (n < 12800)      { l = 0; p = n;         w = 128; s = 8;   hw = 12800; }
  else if (n < 16000) { l = 1; p = n - 12800; w = 64;  s = 16;  hw = 3200;  }
  else if (n < 16800) { l = 2; p = n - 16000; w = 32;  s = 32;  hw = 800;   }
  else if (n < 17008) { l = 3; p = n - 16800; w = 16;  s = 64;  hw = 208;   }
  else                { l = 4; p = n - 17008; w = 8;   s = 128; hw = 56;    }

  int y = p / w, x = p - y * w;
  float cx = (float)(x * s + (s >> 1));
  float cy = (float)(y * s + (s >> 1));

  const float* cp = f.cls[l] + (size_t)b * C_ * hw + p;
  const float* rp = f.reg[l] + (size_t)b * 4 * hw + p;
  const float* tp = f.ctr[l] + (size_t)b * hw + p;
  __builtin_prefetch(rp, 0, 0);   // global_prefetch on gfx1250
  __builtin_prefetch(tp, 0, 0);

  float m = cp[0];
  int   am = 0;
  for (int c = 1; c < C_; ++c) {
    float v = cp[(size_t)c * hw];
    if (v > m) { m = v; am = c; }     // strict '>' keeps first occurrence (jnp.argmax)
  }
  float score = sigmoidf_(m) * sigmoidf_(*tp);

  float rl = rp[0];
  float rt = rp[(size_t)hw];
  float rr = rp[(size_t)2 * hw];
  float rb = rp[(size_t)3 * hw];

  size_t bi = ((size_t)b * NANCH + n) * 4;
  boxes_all[bi + 0] = cx - rl;
  boxes_all[bi + 1] = cy - rt;
  boxes_all[bi + 2] = cx + rr;
  boxes_all[bi + 3] = cy + rb;
  cls_all[(size_t)b * NANCH + n] = am + 1;

  kb[n] = ((uint64_t)fkey(score) << 32) | (uint32_t)(~(uint32_t)n);
}

// ---------------------------------------------------------------------------
// Bitonic sort helpers. Direction: descending iff (index_in_batch & k) == 0.
// ---------------------------------------------------------------------------
__device__ __forceinline__ void lds_cmpxchg(uint64_t* lk, int i, int j, bool desc) {
  uint64_t a = lk[i], b = lk[j];
  bool sw = desc ? (a < b) : (a > b);
  if (sw) { lk[i] = b; lk[j] = a; }
}

// Stage CHUNK keys to LDS, fully sort (k=2..CHUNK), write back.
__global__ void __launch_bounds__(1024)
k_sort_local(uint64_t* __restrict__ keys) {
  __shared__ __align__(16) uint64_t lk[CHUNK];
  int t = threadIdx.x;                         // 1024 threads, 2 keys each
  size_t base = (size_t)blockIdx.x * CHUNK;
  int ib_base = (int)(base & (SORTN - 1));     // index-in-batch of chunk start

  async_g2l_b128(&lk[2 * t], keys + base + 2 * t);
  wait_async0();
  __syncthreads();

  for (int k = 2; k <= CHUNK; k <<= 1) {
    for (int j = k >> 1; j > 0; j >>= 1) {
      int i = ((t & ~(j - 1)) << 1) | (t & (j - 1));   // bit j clear
      bool desc = (((ib_base + i) & k) == 0);
      lds_cmpxchg(lk, i, i | j, desc);
      __syncthreads();
    }
  }

  async_l2g_b128(keys + base + 2 * t, &lk[2 * t]);
  wait_async0();
}

// Global compare-exchange pass (j >= CHUNK). One thread per pair.
__global__ void k_bitonic_global(uint64_t* __restrict__ keys, int k, int j) {
  int t = blockIdx.x * blockDim.x + threadIdx.x;
  if (t >= B_ * (SORTN / 2)) return;
  int b = t / (SORTN / 2);
  int u = t - b * (SORTN / 2);
  int i = ((u & ~(j - 1)) << 1) | (u & (j - 1));
  int ix = i | j;
  uint64_t* kb = keys + (size_t)b * SORTN;
  uint64_t a = kb[i], c = kb[ix];
  bool desc = ((i & k) == 0);
  bool sw = desc ? (a < c) : (a > c);
  if (sw) { kb[i] = c; kb[ix] = a; }
}

// Tail of a k-merge: all passes with j <= CHUNK/2 done in LDS.
__global__ void __launch_bounds__(1024)
k_merge_local(uint64_t* __restrict__ keys, int k) {
  __shared__ __align__(16) uint64_t lk[CHUNK];
  int t = threadIdx.x;
  size_t base = (size_t)blockIdx.x * CHUNK;
  int ib_base = (int)(base & (SORTN - 1));

  async_g2l_b128(&lk[2 * t], keys + base + 2 * t);
  wait_async0();
  __syncthreads();

  for (int j = CHUNK >> 1; j > 0; j >>= 1) {
    int i = ((t & ~(j - 1)) << 1) | (t & (j - 1));
    bool desc = (((ib_base + i) & k) == 0);
    lds_cmpxchg(lk, i, i | j, desc);
    __syncthreads();
  }

  async_l2g_b128(keys + base + 2 * t, &lk[2 * t]);
  wait_async0();
}

// ---------------------------------------------------------------------------
// Kernel 3: gather top-1000, per-batch max_coord reduction, class-offset boxes.
// ---------------------------------------------------------------------------
__global__ void __launch_bounds__(1024)
k_gather(const uint64_t* __restrict__ keys, const float* __restrict__ boxes_all,
         const int* __restrict__ cls_all, float* __restrict__ scores_top,
         int* __restrict__ classes_top, float* __restrict__ boxes_top,
         float* __restrict__ boxes_nms) {
  int b = blockIdx.x;
  int r = threadIdx.x;
  __shared__ float red[1024];

  float bx0 = 0.f, bx1 = 0.f, bx2 = 0.f, bx3 = 0.f;
  float score = 0.f;
  int   cls = 0;
  float lmax = -3.4e38f;
  if (r < NTOP) {
    uint64_t key = keys[(size_t)b * SORTN + r];
    uint32_t n = ~(uint32_t)(key & 0xFFFFFFFFull);
    score = unfkey((uint32_t)(key >> 32));
    cls = cls_all[(size_t)b * NANCH + n];
    size_t bi = ((size_t)b * NANCH + n) * 4;
    bx0 = boxes_all[bi + 0];
    bx1 = boxes_all[bi + 1];
    bx2 = boxes_all[bi + 2];
    bx3 = boxes_all[bi + 3];
    lmax = fmaxf(fmaxf(bx0, bx1), fmaxf(bx2, bx3));
  }
  red[r] = lmax;
  __syncthreads();
  for (int off = 512; off > 0; off >>= 1) {
    if (r < off) red[r] = fmaxf(red[r], red[r + off]);
    __syncthreads();
  }
  float maxc = red[0];

  if (r < NTOP) {
    size_t o = (size_t)b * NTOP + r;
    scores_top[o] = score;
    classes_top[o] = cls;
    float offv = (float)cls * (maxc + 1.0f);
    boxes_top[o * 4 + 0] = bx0;
    boxes_top[o * 4 + 1] = bx1;
    boxes_top[o * 4 + 2] = bx2;
    boxes_top[o * 4 + 3] = bx3;
    boxes_nms[o * 4 + 0] = bx0 + offv;
    boxes_nms[o * 4 + 1] = bx1 + offv;
    boxes_nms[o * 4 + 2] = bx2 + offv;
    boxes_nms[o * 4 + 3] = bx3 + offv;
  }
}

// ---------------------------------------------------------------------------
// Kernel 4: sequential NMS; boxes staged into LDS by the Tensor Data Mover
// (one tensor_load_to_lds per block, TENSORcnt), fallback = per-lane async.
// ---------------------------------------------------------------------------
__global__ void __launch_bounds__(1024)
k_nms(const float* __restrict__ boxes_nms, const float* __restrict__ scores_top,
      int* __restrict__ keep) {
  int b = blockIdx.x;
  int tid = threadIdx.x;
  __shared__ __align__(16) float lbox[NTOP * 4];   // 16000 B
  __shared__ unsigned char sup_s[NTOP];            // 1000 B

#if __has_builtin(__builtin_amdgcn_tensor_load_to_lds)
  if ((tid >> 5) == 0) {                 // wave-uniform guard: only wave 0 issues DMA
    const float* gsrc = boxes_nms + (size_t)b * NTOP * 4;
    uint64_t ga = (uint64_t)(uintptr_t)gsrc;
    uint32_t lds = (uint32_t)(uintptr_t)&lbox[0];
    // D# group0: count=1, lds_addr, 57-bit global_addr, type=2 (ISA 8.3)
    u32x4 g0;
    g0.x = 1u;                                   // count=1, user mode
    g0.y = lds;                                  // lds_addr
    g0.z = (uint32_t)(ga & 0xFFFFFFFFull);       // global_addr[31:0]
    g0.w = (uint32_t)((ga >> 32) & 0x01FFFFFFull) | (2u << 30);  // addr[56:32] | type=2
    // D# group1: data_size=4B, tensor 4000x1, tile 4000x1, stride0=4000 (ISA 8.4)
    i32x8 g1;
    g1.s0 = (int)(2u << 16);                     // workgroup_mask=0, data_size=2 (4B)
    g1.s1 = (int)((4000u & 0xFFFFu) << 16);      // tensor_dim0[15:0] at bits 63:48
    g1.s2 = (int)(1u << 16);                     // tensor_dim0 hi=0 | tensor_dim1=1
    g1.s3 = (int)(4000u << 16);                  // tensor_dim1 hi=0 | tile_dim0=4000
    g1.s4 = 1;                                   // tile_dim1=1, tile_dim2=0
    g1.s5 = 4000;                                // tensor_dim0_stride lo32
    g1.s6 = 0;                                   // stride0 hi | stride1 lo
    g1.s7 = 0;
    i32x4 gz = {0, 0, 0, 0};
#if __clang_major__ >= 23
    i32x8 gz8 = {0, 0, 0, 0, 0, 0, 0, 0};
    __builtin_amdgcn_tensor_load_to_lds(g0, g1, gz, gz, gz8, 0);
#else
    __builtin_amdgcn_tensor_load_to_lds(g0, g1, gz, gz, 0);
#endif
    __builtin_amdgcn_s_wait_tensorcnt(0);
  }
#else
  if (tid < NTOP) {
    async_g2l_b128(&lbox[tid * 4], boxes_nms + ((size_t)b * NTOP + tid) * 4);
  }
  wait_async0();
#endif
  if (tid < NTOP) {
    sup_s[tid] = (scores_top[(size_t)b * NTOP + tid] >= SCORE_THR) ? 0 : 1;
  }
  __syncthreads();

  bool mine = (tid < NTOP);
  float bx0 = 0.f, bx1 = 0.f, bx2 = 0.f, bx3 = 0.f, aj = 1.f;
  if (mine) {
    bx0 = lbox[tid * 4 + 0];
    bx1 = lbox[tid * 4 + 1];
    bx2 = lbox[tid * 4 + 2];
    bx3 = lbox[tid * 4 + 3];
    aj = (bx2 - bx0 + 1.0f) * (bx3 - bx1 + 1.0f);
  }

  for (int i = 0; i < NTOP; ++i) {
    __syncthreads();
    if (sup_s[i]) continue;                  // uniform branch across the block
    if (mine && tid > i && !sup_s[tid]) {
      float x1i = lbox[i * 4 + 0];
      float y1i = lbox[i * 4 + 1];
      float x2i = lbox[i * 4 + 2];
      float y2i = lbox[i * 4 + 3];
      float ai  = (x2i - x1i + 1.0f) * (y2i - y1i + 1.0f);
      float iw  = fmaxf(fminf(x2i, bx2) - fmaxf(x1i, bx0), 0.0f);
      float ih  = fmaxf(fminf(y2i, bx3) - fmaxf(y1i, bx1), 0.0f);
      float inter = iw * ih;
      float iou = inter / (ai + aj - inter);
      if (iou > IOU_THR) sup_s[tid] = 1;
    }
  }
  __syncthreads();
  if (mine) keep[(size_t)b * NTOP + tid] = sup_s[tid] ? 0 : 1;
}

// ---------------------------------------------------------------------------
// Kernel 5: masked outputs, concatenated flat: scores | classes | boxes.
// ---------------------------------------------------------------------------
__global__ void k_final(const float* __restrict__ scores_top,
                        const int* __restrict__ classes_top,
                        const float* __restrict__ boxes_top,
                        const int* __restrict__ keep, float* __restrict__ out) {
  int t = blockIdx.x * blockDim.x + threadIdx.x;
  if (t >= B_ * NTOP) return;
  int k = keep[t];
  float kf = (float)k;
  out[t] = scores_top[t] * kf;
  out[B_ * NTOP + t] = (float)(classes_top[t] * k);
  float* ob = out + 2 * B_ * NTOP;
  ob[t * 4 + 0] = boxes_top[t * 4 + 0] * kf;
  ob[t * 4 + 1] = boxes_top[t * 4 + 1] * kf;
  ob[t * 4 + 2] = boxes_top[t * 4 + 2] * kf;
  ob[t * 4 + 3] = boxes_top[t * 4 + 3] * kf;
}

extern "C" void kernel_launch(void* const* d_in, const int* in_sizes, int n_in,
                              void* d_out, int out_size, void* d_ws, size_t ws_size,
                              hipStream_t stream) {
  (void)in_sizes; (void)n_in; (void)out_size; (void)ws_size;

  Feats f;
  for (int l = 0; l < 5; ++l) {
    f.cls[l] = (const float*)d_in[l];
    f.ctr[l] = (const float*)d_in[5 + l];
    f.reg[l] = (const float*)d_in[10 + l];
  }

  // workspace layout (256B-aligned slabs)
  char* ws = (char*)d_ws;
  size_t off = 0;
  auto take = [&](size_t bytes) -> char* {
    char* p = ws + off;
    off = (off + bytes + 255) & ~(size_t)255;
    return p;
  };
  uint64_t* keys       = (uint64_t*)take((size_t)B_ * SORTN * 8);
  float*    boxes_all  = (float*)   take((size_t)B_ * NANCH * 4 * 4);
  int*      cls_all    = (int*)     take((size_t)B_ * NANCH * 4);
  float*    scores_top = (float*)   take((size_t)B_ * NTOP * 4);
  int*      classes_top= (int*)     take((size_t)B_ * NTOP * 4);
  float*    boxes_top  = (float*)   take((size_t)B_ * NTOP * 4 * 4);
  float*    boxes_nms  = (float*)   take((size_t)B_ * NTOP * 4 * 4);
  int*      keep       = (int*)     take((size_t)B_ * NTOP * 4);

  const int threads = 256;
  const int total = B_ * SORTN;
  const int nchunks = total / CHUNK;           // 256 blocks of 1024 threads

  k_score<<<(total + threads - 1) / threads, threads, 0, stream>>>(f, keys, boxes_all, cls_all);

  // Bitonic sort: local full sort of 2048-chunks, then global+local merge tail.
  k_sort_local<<<nchunks, 1024, 0, stream>>>(keys);
  for (int k = 2 * CHUNK; k <= SORTN; k <<= 1) {
    for (int j = k >> 1; j >= CHUNK; j >>= 1)
      k_bitonic_global<<<(total / 2 + threads - 1) / threads, threads, 0, stream>>>(keys, k, j);
    k_merge_local<<<nchunks, 1024, 0, stream>>>(keys, k);
  }

  k_gather<<<B_, 1024, 0, stream>>>(keys, boxes_all, cls_all, scores_top, classes_top,
                                    boxes_top, boxes_nms);
  k_nms<<<B_, 1024, 0, stream>>>(boxes_nms, scores_top, keep);
  k_final<<<(B_ * NTOP + threads - 1) / threads, threads, 0, stream>>>(
      scores_top, classes_top, boxes_top, keep, (float*)d_out);
}